// VariationalGCNEncoder_7060926234910
// MI455X (gfx1250) — compile-verified
//
#include <hip/hip_runtime.h>

// ---------------------------------------------------------------------------
// VariationalGCNEncoder for MI455X (gfx1250, wave32)
//
// Dense GEMMs use V_WMMA_F32_16X16X4_F32 (exact fp32 matrix core path).
// Edge aggregation uses wave-per-edge gather + global_atomic_add_f32 scatter;
// feature matrices (25.6 MB) are L2-resident on the 192 MB L2, so the
// scatter phase is L2-atomic bound, not HBM bound.
// ---------------------------------------------------------------------------

#define N_NODES 50000
#define N_EDGES 600000
#define IN_CH   128
#define HID     128   // layer-1 output width; also mu(64)+logstd(64) fused
#define OUT_CH  64

typedef __attribute__((ext_vector_type(2))) float v2f;
typedef __attribute__((ext_vector_type(8))) float v8f;

// ---- degree accumulation: deg[col] += w_eff --------------------------------
__global__ void deg_kernel(const int* __restrict__ row, const int* __restrict__ col,
                           const float* __restrict__ w, float* __restrict__ deg) {
    int e = blockIdx.x * blockDim.x + threadIdx.x;
    if (e >= N_EDGES) return;
    int r = row[e], c = col[e];
    float we = (r == c) ? 0.0f : w[e];   // add_remaining_self_loops drops existing self-loops
    if (we != 0.0f) atomicAdd(&deg[c], we);
}

// ---- per-node: deg+=1 (fresh self-loop), dis=rsqrt(deg), invdeg=1/deg ------
__global__ void dis_kernel(const float* __restrict__ deg, float* __restrict__ dis,
                           float* __restrict__ invdeg) {
    int i = blockIdx.x * blockDim.x + threadIdx.x;
    if (i >= N_NODES) return;
    float d = deg[i] + 1.0f;          // deg >= 1, rsqrt safe
    dis[i]    = rsqrtf(d);
    invdeg[i] = 1.0f / d;
}

// ---- C[M,128] = X[M,128] @ Wcat^T via V_WMMA_F32_16X16X4_F32 ---------------
// Wcat row r = Wa[r] if r < n_split else Wb[r - n_split]  (both row-major [*,128]).
// One wave computes one 16x16 C tile; K=128 chained through 32 WMMAs.
//
// VGPR layouts per CDNA5 ISA 7.12.2 (fp32, wave32):
//   A 16x4 : lane<16 -> (M=lane, K=k+{0,1}); lane>=16 -> (M=lane-16, K=k+{2,3})
//   B 4x16 : lane<16 -> (K=k+{0,1}, N=lane); lane>=16 -> (K=k+{2,3}, N=lane-16)
//   C/D    : VGPR j  -> (M = j + 8*(lane>=16), N = lane&15)
__global__ void gemm_xwt_wmma(const float* __restrict__ X,
                              const float* __restrict__ Wa,
                              const float* __restrict__ Wb,
                              int n_split,
                              float* __restrict__ C) {
    int w     = blockIdx.x * 8 + (threadIdx.x >> 5);   // 8 waves per block
    int lane  = threadIdx.x & 31;
    int nTile = w & 7;            // 8 tiles of 16 cover the 128 output cols
    int mTile = w >> 3;
    if (mTile >= N_NODES / 16) return;

    int half = lane >> 4;         // 0: K low pair, 1: K high pair
    int l16  = lane & 15;

    const float* xrow = X + (size_t)(mTile * 16 + l16) * IN_CH + half * 2;
    int ng = nTile * 16 + l16;
    const float* wrow = ((ng < n_split) ? (Wa + (size_t)ng * IN_CH)
                                        : (Wb + (size_t)(ng - n_split) * IN_CH)) + half * 2;

    v8f acc = {};
#pragma unroll
    for (int k = 0; k < IN_CH; k += 4) {
        v2f a = *(const v2f*)(xrow + k);   // A fragment: X[m, k..k+3]
        v2f b = *(const v2f*)(wrow + k);   // B fragment: W^T[k..k+3, n] = W[n, k..k+3]
        // 8 args: (neg_a, A, neg_b, B, c_mod, C, reuse_a, reuse_b)
        acc = __builtin_amdgcn_wmma_f32_16x16x4_f32(
            false, a, false, b, (short)0, acc, false, false);
    }

    float* crow = C + (size_t)(mTile * 16 + half * 8) * HID + nTile * 16 + l16;
#pragma unroll
    for (int j = 0; j < 8; ++j) crow[(size_t)j * HID] = acc[j];
}

// ---- edge scatter: agg[col] += dis[row]*w_eff*dis[col] * src[row] ----------
// One wave per edge; each lane handles 4 of the 128 channels (float4 gather,
// 4x global_atomic_add_f32 scatter).
__global__ void scatter_kernel(const int* __restrict__ row, const int* __restrict__ col,
                               const float* __restrict__ w, const float* __restrict__ dis,
                               const float* __restrict__ src, float* __restrict__ agg) {
    int wave = (blockIdx.x * blockDim.x + threadIdx.x) >> 5;
    int lane = threadIdx.x & 31;
    if (wave >= N_EDGES) return;
    int r = row[wave], c = col[wave];
    float we = (r == c) ? 0.0f : w[wave];
    if (we == 0.0f) return;                       // self-loops contribute nothing here
    float coeff = dis[r] * we * dis[c];

    float4 v = ((const float4*)(src + (size_t)r * HID))[lane];
    float* dst = agg + (size_t)c * HID + lane * 4;
    atomicAdd(dst + 0, coeff * v.x);
    atomicAdd(dst + 1, coeff * v.y);
    atomicAdd(dst + 2, coeff * v.z);
    atomicAdd(dst + 3, coeff * v.w);
}

// ---- h = relu(agg1 + xw*invdeg + b1), written in place over agg1 -----------
__global__ void act_kernel(float* __restrict__ agg, const float* __restrict__ xw,
                           const float* __restrict__ invdeg, const float* __restrict__ b1) {
    int idx = blockIdx.x * blockDim.x + threadIdx.x;
    if (idx >= N_NODES * HID) return;
    int node = idx >> 7;      // /128
    int ch   = idx & 127;
    float v = agg[idx] + xw[idx] * invdeg[node] + b1[ch];
    agg[idx] = fmaxf(v, 0.0f);
}

// ---- final: split fused [mu|logstd] channels, add self-loop term + bias ----
__global__ void out_kernel(const float* __restrict__ agg2, const float* __restrict__ xw2,
                           const float* __restrict__ invdeg, const float* __restrict__ b_mu,
                           const float* __restrict__ b_ls, float* __restrict__ out) {
    int idx = blockIdx.x * blockDim.x + threadIdx.x;
    if (idx >= N_NODES * OUT_CH) return;
    int node = idx >> 6;      // /64
    int ch   = idx & 63;
    size_t base = (size_t)node * HID;
    float id = invdeg[node];
    out[idx] = agg2[base + ch] + xw2[base + ch] * id + b_mu[ch];
    out[(size_t)N_NODES * OUT_CH + idx] =
        agg2[base + OUT_CH + ch] + xw2[base + OUT_CH + ch] * id + b_ls[ch];
}

extern "C" void kernel_launch(void* const* d_in, const int* in_sizes, int n_in,
                              void* d_out, int out_size, void* d_ws, size_t ws_size,
                              hipStream_t stream) {
    (void)in_sizes; (void)n_in; (void)out_size; (void)ws_size;
    const float* x   = (const float*)d_in[0];
    const int*   ei  = (const int*)d_in[1];
    const float* ew  = (const float*)d_in[2];
    const float* W1  = (const float*)d_in[3];
    const float* b1  = (const float*)d_in[4];
    const float* Wmu = (const float*)d_in[5];
    const float* bmu = (const float*)d_in[6];
    const float* Wls = (const float*)d_in[7];
    const float* bls = (const float*)d_in[8];
    const int* row = ei;
    const int* col = ei + N_EDGES;

    // workspace layout (floats); all float4-aligned offsets
    float* ws     = (float*)d_ws;
    float* dis    = ws;                          // 50000
    float* invdeg = ws + 50000;                  // 50000
    float* deg    = ws + 100000;                 // 50000
    float* xw     = ws + 150000;                 // 6,400,000 (layer1 xw, reused for layer2)
    float* agg1   = ws + 150000 + 6400000;       // 6,400,000 (becomes h in place)
    float* agg2   = ws + 150000 + 12800000;      // 6,400,000
    // total = 19,350,000 floats ~ 77.4 MB

    // zero accumulators every call (ws is not re-poisoned between replays)
    hipMemsetAsync(deg,  0, (size_t)N_NODES * sizeof(float), stream);
    hipMemsetAsync(agg1, 0, (size_t)N_NODES * HID * sizeof(float), stream);
    hipMemsetAsync(agg2, 0, (size_t)N_NODES * HID * sizeof(float), stream);

    // normalization
    deg_kernel<<<(N_EDGES + 255) / 256, 256, 0, stream>>>(row, col, ew, deg);
    dis_kernel<<<(N_NODES + 255) / 256, 256, 0, stream>>>(deg, dis, invdeg);

    // layer 1: xw = x @ W1^T   (3125 m-tiles * 8 n-tiles = 25000 waves)
    gemm_xwt_wmma<<<3125, 256, 0, stream>>>(x, W1, W1, HID, xw);
    scatter_kernel<<<N_EDGES / 8, 256, 0, stream>>>(row, col, ew, dis, xw, agg1);
    act_kernel<<<(N_NODES * HID + 255) / 256, 256, 0, stream>>>(agg1, xw, invdeg, b1);

    // layer 2 (mu & logstd fused): xw2 = h @ [W_mu;W_ls]^T, overwrites xw buffer
    gemm_xwt_wmma<<<3125, 256, 0, stream>>>(agg1, Wmu, Wls, OUT_CH, xw);
    scatter_kernel<<<N_EDGES / 8, 256, 0, stream>>>(row, col, ew, dis, xw, agg2);

    out_kernel<<<(N_NODES * OUT_CH + 255) / 256, 256, 0, stream>>>(
        agg2, xw, invdeg, bmu, bls, (float*)d_out);
}